// GCN_R_23192823399150
// MI455X (gfx1250) — compile-verified
//
#include <hip/hip_runtime.h>
#include <math.h>

typedef float v2f __attribute__((ext_vector_type(2)));
typedef float v8f __attribute__((ext_vector_type(8)));

#define N_NODES 50000
#define N_EDGES 600000
#define DFEAT   128
#define NCLS    40

// ---------------------------------------------------------------------------
// Degree / normalization
// ---------------------------------------------------------------------------
__global__ void k_init_deg(int* deg, int n) {
    int i = blockIdx.x * blockDim.x + threadIdx.x;
    if (i < n) deg[i] = 1;  // self-loop
}

__global__ void k_count_deg(int* deg, const int* __restrict__ dst, int e) {
    int i = blockIdx.x * blockDim.x + threadIdx.x;
    if (i < e) atomicAdd(&deg[dst[i]], 1);
}

__global__ void k_make_dinv(float* dinv, int n) {
    int i = blockIdx.x * blockDim.x + threadIdx.x;
    if (i < n) {
        int d = ((const int*)dinv)[i];
        dinv[i] = rsqrtf((float)d);   // d >= 1 always (self-loop)
    }
}

__global__ void k_zero(float* p, int n) {
    int i = blockIdx.x * blockDim.x + threadIdx.x;
    if (i < n) p[i] = 0.0f;
}

// ---------------------------------------------------------------------------
// GEMM: C[M x N] = A[M x 128] @ B[128 x N].  N is compile-time so every
// in-loop access is base + 32-bit immediate offset (no 64-bit muls).
// One wave per 16x16 output tile, 32 chained v_wmma_f32_16x16x4_f32.
// grid: (M/16, ceil(N/16)/wavesPerBlock), block: 32*wavesPerBlock
// ---------------------------------------------------------------------------
template<int N>
__global__ void k_gemm(const float* __restrict__ A, const float* __restrict__ B,
                       float* __restrict__ C) {
    const int lane  = threadIdx.x & 31;
    const int wave  = threadIdx.x >> 5;
    const int row0  = blockIdx.x * 16;
    const int col0  = (blockIdx.y * (blockDim.x >> 5) + wave) * 16;
    const int r     = lane & 15;          // A row in tile; B/C col in tile
    const int kb    = (lane >> 4) * 2;    // K sub-offset (0 or 2)

    // Clamp column so every lane's loads stay in-bounds even when N is not a
    // multiple of 16 (head: N=40, padded to 48). EXEC stays all-ones for WMMA;
    // out-of-range lanes just compute garbage that is dropped at the store.
    const int col  = col0 + r;
    const int colc = (col < N) ? col : (N - 1);

    const float* Ap = A + (row0 + r) * DFEAT + kb;   // + k0        per step
    const float* Bp = B + kb * N + colc;             // + k0*N, +N  per step

    v8f acc = {};
    #pragma unroll
    for (int k0 = 0; k0 < DFEAT; k0 += 4) {
        v2f a = *(const v2f*)(Ap + k0);              // contiguous K pair
        v2f b;
        b.x = Bp[k0 * N];
        b.y = Bp[k0 * N + N];
        acc = __builtin_amdgcn_wmma_f32_16x16x4_f32(
            false, a, false, b, (short)0, acc, false, false);
    }

    if (col < N) {
        // D layout: VGPR i -> row (i + 8*(lane>>4)), col r
        float* Cp = C + (row0 + (lane >> 4) * 8) * N + col;
        #pragma unroll
        for (int i = 0; i < 8; ++i)
            Cp[i * N] = acc[i];
    }
}

// Head variant: adds the class bias at the store.
template<int N>
__global__ void k_gemm_bias(const float* __restrict__ A, const float* __restrict__ B,
                            const float* __restrict__ bias, float* __restrict__ C) {
    const int lane  = threadIdx.x & 31;
    const int wave  = threadIdx.x >> 5;
    const int row0  = blockIdx.x * 16;
    const int col0  = wave * 16;
    const int r     = lane & 15;
    const int kb    = (lane >> 4) * 2;

    const int col  = col0 + r;
    const int colc = (col < N) ? col : (N - 1);

    const float* Ap = A + (row0 + r) * DFEAT + kb;
    const float* Bp = B + kb * N + colc;

    v8f acc = {};
    #pragma unroll
    for (int k0 = 0; k0 < DFEAT; k0 += 4) {
        v2f a = *(const v2f*)(Ap + k0);
        v2f b;
        b.x = Bp[k0 * N];
        b.y = Bp[k0 * N + N];
        acc = __builtin_amdgcn_wmma_f32_16x16x4_f32(
            false, a, false, b, (short)0, acc, false, false);
    }

    if (col < N) {
        const float bb = bias[col];
        float* Cp = C + (row0 + (lane >> 4) * 8) * N + col;
        #pragma unroll
        for (int i = 0; i < 8; ++i)
            Cp[i * N] = acc[i] + bb;
    }
}

// ---------------------------------------------------------------------------
// Edge scatter: agg[dst] += h[src] * dinv[src]*dinv[dst]
// One wave per edge; each lane moves 4 contiguous floats (128 total).
// ---------------------------------------------------------------------------
__global__ void k_scatter(const float* __restrict__ h, const int* __restrict__ src,
                          const int* __restrict__ dst, const float* __restrict__ dinv,
                          float* __restrict__ agg, int E) {
    const int e = blockIdx.x * (blockDim.x >> 5) + (threadIdx.x >> 5);
    if (e >= E) return;
    const int lane = threadIdx.x & 31;
    const int s = src[e];
    const int d = dst[e];
    const float norm = dinv[s] * dinv[d];
    const float4 v = *(const float4*)(h + s * DFEAT + lane * 4);
    float* out = agg + d * DFEAT + lane * 4;
    unsafeAtomicAdd(out + 0, v.x * norm);
    unsafeAtomicAdd(out + 1, v.y * norm);
    unsafeAtomicAdd(out + 2, v.z * norm);
    unsafeAtomicAdd(out + 3, v.w * norm);
}

// ---------------------------------------------------------------------------
// Finalize: out = tanh(agg + h*dinv^2 + b)   (self-loop term + bias + act)
// Safe in-place on agg.
// ---------------------------------------------------------------------------
__global__ void k_finalize(const float* __restrict__ h, const float* __restrict__ agg,
                           const float* __restrict__ dinv, const float* __restrict__ b,
                           float* __restrict__ out, int n) {
    int t = blockIdx.x * blockDim.x + threadIdx.x;
    if (t >= n) return;
    const int i = t >> 7;      // node
    const int f = t & 127;     // feature
    const float di = dinv[i];
    out[t] = tanhf(agg[t] + h[t] * di * di + b[f]);
}

// ---------------------------------------------------------------------------
extern "C" void kernel_launch(void* const* d_in, const int* in_sizes, int n_in,
                              void* d_out, int out_size, void* d_ws, size_t ws_size,
                              hipStream_t stream) {
    const float* x    = (const float*)d_in[0];
    const int*   ei   = (const int*)  d_in[1];   // [2, E] flat: src then dst
    const float* W1   = (const float*)d_in[2];
    const float* b1   = (const float*)d_in[3];
    const float* W2   = (const float*)d_in[4];
    const float* b2   = (const float*)d_in[5];
    const float* Wlin = (const float*)d_in[6];
    const float* blin = (const float*)d_in[7];

    const int* src = ei;
    const int* dst = ei + N_EDGES;

    float* wsf  = (float*)d_ws;
    float* dinv = wsf;                         // N_NODES floats (ints first)
    float* bufA = wsf + 50048;                 // N*128
    float* bufB = bufA + (size_t)N_NODES * DFEAT;

    const int NF = N_NODES * DFEAT;
    const dim3 gemmGrid(N_NODES / 16, DFEAT / 64);   // 3125 x 2, 4 waves/block

    // normalization coefficients
    k_init_deg <<<(N_NODES + 255) / 256, 256, 0, stream>>>((int*)dinv, N_NODES);
    k_count_deg<<<(N_EDGES + 255) / 256, 256, 0, stream>>>((int*)dinv, dst, N_EDGES);
    k_make_dinv<<<(N_NODES + 255) / 256, 256, 0, stream>>>(dinv, N_NODES);

    // layer 1: h1 = tanh(Ahat (x W1) + b1)
    k_gemm<DFEAT><<<gemmGrid, 128, 0, stream>>>(x, W1, bufA);
    k_zero    <<<(NF + 255) / 256, 256, 0, stream>>>(bufB, NF);
    k_scatter <<<(N_EDGES + 7) / 8, 256, 0, stream>>>(bufA, src, dst, dinv, bufB, N_EDGES);
    k_finalize<<<(NF + 255) / 256, 256, 0, stream>>>(bufA, bufB, dinv, b1, bufB, NF);

    // layer 2: h2 = tanh(Ahat (h1 W2) + b2)
    k_gemm<DFEAT><<<gemmGrid, 128, 0, stream>>>(bufB, W2, bufA);
    k_zero    <<<(NF + 255) / 256, 256, 0, stream>>>(bufB, NF);
    k_scatter <<<(N_EDGES + 7) / 8, 256, 0, stream>>>(bufA, src, dst, dinv, bufB, N_EDGES);
    k_finalize<<<(NF + 255) / 256, 256, 0, stream>>>(bufA, bufB, dinv, b2, bufB, NF);

    // head: out = h2 @ Wlin + blin  (3 waves/block cover 48 cols, 40 valid)
    k_gemm_bias<NCLS><<<N_NODES / 16, 96, 0, stream>>>(bufB, Wlin, blin, (float*)d_out);
}